// AdvancedTripletLoss_52424370815540
// MI455X (gfx1250) — compile-verified
//
#include <hip/hip_runtime.h>

#define BN 768
#define DIM 128
#define MARGIN 1.0f
#define EPSV 1e-12f
#define NTILE (BN / 16)

typedef float v2f __attribute__((ext_vector_type(2)));
typedef float v8f __attribute__((ext_vector_type(8)));

// ---------------- Kernel 0: squared row norms, one wave per row ----------------
__global__ void row_norms_kernel(const float* __restrict__ E, float* __restrict__ sq) {
    const int row  = blockIdx.x;
    const int lane = threadIdx.x;                 // 32 threads (wave32)
    const float* r = E + row * DIM;
    float acc = 0.0f;
#pragma unroll
    for (int k = lane; k < DIM; k += 32) acc += r[k] * r[k];
#pragma unroll
    for (int off = 16; off > 0; off >>= 1) acc += __shfl_xor(acc, off, 32);
    if (lane == 0) sq[row] = acc;
}

// ------- Kernel 1: d[i][j] = sqrt(max(sq_i + sq_j - 2*E_i.E_j, eps)) via WMMA -------
// One wave per 16x16 output tile. K=128 in steps of 4 using V_WMMA_F32_16X16X4_F32.
// A fragment (16x4 f32): lane L -> M = L&15, VGPR v holds K = v + 2*(L>>4).
// B fragment (4x16 f32): mirrored with N per lane; since B-tile = rows j0..j0+15 of E
// (computing E*E^T), the gather pattern is identical to A.
__global__ void dist_wmma_kernel(const float* __restrict__ E,
                                 const float* __restrict__ sq,
                                 float* __restrict__ dmat) {
    const int tile = blockIdx.x;
    const int i0 = (tile / NTILE) * 16;
    const int j0 = (tile % NTILE) * 16;
    const int lane = threadIdx.x;                 // 32 threads
    const int mr   = lane & 15;
    const int kb   = (lane >> 4) * 2;             // lanes 16-31 hold K+2,K+3

    const float* arow = E + (i0 + mr) * DIM + kb;
    const float* brow = E + (j0 + mr) * DIM + kb;

    v8f c = {};
#pragma unroll
    for (int k0 = 0; k0 < DIM; k0 += 4) {
        v2f a = *(const v2f*)(arow + k0);
        v2f b = *(const v2f*)(brow + k0);
        // (neg_a, A, neg_b, B, c_mod, C, reuse_a, reuse_b)
        c = __builtin_amdgcn_wmma_f32_16x16x4_f32(false, a, false, b,
                                                  (short)0, c, false, false);
    }

    const int half = lane >> 4;
    const int col  = j0 + (lane & 15);
    const float sqj = sq[col];
#pragma unroll
    for (int r = 0; r < 8; ++r) {
        const int row = i0 + r + 8 * half;        // C/D layout: VGPR r <-> M=r+8*half
        const float d2 = sq[row] + sqj - 2.0f * c[r];
        dmat[row * BN + col] = __builtin_sqrtf(fmaxf(d2, EPSV));
    }
}

// ---- Kernel 2: per-anchor semi-hard mining + partial sums (block per anchor) ----
__global__ void triplet_rows_kernel(const float* __restrict__ dmat,
                                    const int* __restrict__ labels,
                                    float* __restrict__ psum,
                                    float* __restrict__ pcnt) {
    __shared__ float s_d[BN];
    __shared__ int   s_lab[BN];
    __shared__ float s_red[256];

    const int i   = blockIdx.x;
    const int tid = threadIdx.x;                  // 256 threads = 8 waves

    for (int t = tid; t < BN; t += 256) {
        s_d[t]   = dmat[i * BN + t];
        s_lab[t] = labels[t];
    }
    __syncthreads();

    const int li = s_lab[i];

    // hardest negative distance = min over labels != li (inf if none)
    float mn = __builtin_inff();
    for (int t = tid; t < BN; t += 256)
        if (s_lab[t] != li) mn = fminf(mn, s_d[t]);
    s_red[tid] = mn;
    __syncthreads();
    for (int s = 128; s > 0; s >>= 1) {
        if (tid < s) s_red[tid] = fminf(s_red[tid], s_red[tid + s]);
        __syncthreads();
    }
    const float hardest    = s_red[0];
    const bool  neg_exists = (hardest < __builtin_inff());
    __syncthreads();

    float accs = 0.0f, accc = 0.0f;
    if (neg_exists) {
        for (int p = tid; p < BN; p += 256) {
            if (p == i || s_lab[p] != li) continue;   // pos_mask
            const float dp = s_d[p];
            const float hi = dp + MARGIN;
            float negd = hardest;
            for (int n = 0; n < BN; ++n) {            // first semi-hard index
                const float dn = s_d[n];
                if (s_lab[n] != li && dn > dp && dn < hi) { negd = dn; break; }
            }
            accs += fmaxf(dp - negd + MARGIN, 0.0f);
            accc += 1.0f;
        }
    }

    s_red[tid] = accs;
    __syncthreads();
    for (int s = 128; s > 0; s >>= 1) {
        if (tid < s) s_red[tid] += s_red[tid + s];
        __syncthreads();
    }
    if (tid == 0) psum[i] = s_red[0];
    __syncthreads();

    s_red[tid] = accc;
    __syncthreads();
    for (int s = 128; s > 0; s >>= 1) {
        if (tid < s) s_red[tid] += s_red[tid + s];
        __syncthreads();
    }
    if (tid == 0) pcnt[i] = s_red[0];
}

// ---------------- Kernel 3: deterministic final reduction ----------------
__global__ void finalize_kernel(const float* __restrict__ psum,
                                const float* __restrict__ pcnt,
                                float* __restrict__ out) {
    __shared__ float ss[256];
    __shared__ float sc[256];
    const int tid = threadIdx.x;
    float a = 0.0f, b = 0.0f;
    for (int t = tid; t < BN; t += 256) { a += psum[t]; b += pcnt[t]; }
    ss[tid] = a; sc[tid] = b;
    __syncthreads();
    for (int s = 128; s > 0; s >>= 1) {
        if (tid < s) { ss[tid] += ss[tid + s]; sc[tid] += sc[tid + s]; }
        __syncthreads();
    }
    if (tid == 0) out[0] = ss[0] / fmaxf(sc[0], 1.0f);
}

extern "C" void kernel_launch(void* const* d_in, const int* in_sizes, int n_in,
                              void* d_out, int out_size, void* d_ws, size_t ws_size,
                              hipStream_t stream) {
    const float* E      = (const float*)d_in[0];   // [768,128] f32
    const int*   labels = (const int*)d_in[1];     // [768] i32

    float* ws   = (float*)d_ws;
    float* sq   = ws;                    // 768
    float* dmat = sq + BN;               // 768*768
    float* psum = dmat + BN * BN;        // 768
    float* pcnt = psum + BN;             // 768

    row_norms_kernel<<<BN, 32, 0, stream>>>(E, sq);
    dist_wmma_kernel<<<NTILE * NTILE, 32, 0, stream>>>(E, sq, dmat);
    triplet_rows_kernel<<<BN, 256, 0, stream>>>(dmat, labels, psum, pcnt);
    finalize_kernel<<<1, 256, 0, stream>>>(psum, pcnt, (float*)d_out);
}